// GSAPool_76570676953492
// MI455X (gfx1250) — compile-verified
//
#include <hip/hip_runtime.h>
#include <hip/hip_bf16.h>
#include <math.h>

// ---------------------------------------------------------------------------
// GSAPool pipeline for MI455X (gfx1250, wave32).
// Bandwidth-bound (~450MB total traffic -> ~20us at 23.3 TB/s).
// Score matvec uses V_WMMA_F32_16X16X4_F32: 16 nodes per wave, weight vector
// broadcast across B's 16 columns so every output column equals the dot
// product (B N-layout becomes irrelevant; full f32 accuracy preserved).
// ---------------------------------------------------------------------------

typedef float v2f __attribute__((ext_vector_type(2)));
typedef float v8f __attribute__((ext_vector_type(8)));

#define NBINS 65536

__device__ __forceinline__ unsigned sortkey_desc(float s) {
    unsigned u = __float_as_uint(s);
    // monotonic ascending map for IEEE floats
    u ^= (u & 0x80000000u) ? 0xFFFFFFFFu : 0x80000000u;
    return ~u;  // complement -> ascending key order == descending score order
}

__device__ __forceinline__ float sigmoidf_dev(float v) {
    return 1.0f / (1.0f + __expf(-v));
}

// ---------------------------------------------------------------------------
// 0) init: zero hist + binfill, remap = -1
// ---------------------------------------------------------------------------
__global__ void init_kernel(unsigned* hist, unsigned* binfill, int* remap, int N) {
    int total = 2 * NBINS + N;
    for (int i = blockIdx.x * blockDim.x + threadIdx.x; i < total;
         i += gridDim.x * blockDim.x) {
        if (i < NBINS)               hist[i] = 0u;
        else if (i < 2 * NBINS)      binfill[i - NBINS] = 0u;
        else                         remap[i - 2 * NBINS] = -1;
    }
}

// ---------------------------------------------------------------------------
// 1) scores via WMMA f32 16x16x4.  One wave = 16 nodes.  8 waves / block.
//    acc1 = h_tile x Ws_bcast ; acc2 = oh_tile x Wf_bcast
//    Column 0 of each 16x16 accumulator (lanes 0 and 16) holds the dots.
// ---------------------------------------------------------------------------
__global__ void scores_kernel(const float* __restrict__ h,
                              const float* __restrict__ oh,
                              const float* __restrict__ Ws,
                              const float* __restrict__ Wf,
                              const float* __restrict__ bs_p,
                              const float* __restrict__ bf_p,
                              const float* __restrict__ wv,
                              float* __restrict__ scores,
                              unsigned* __restrict__ keys,
                              unsigned* __restrict__ hist,
                              int N, int D, int ntiles) {
    __shared__ float lws[1024];
    __shared__ float lwf[1024];
    for (int i = threadIdx.x; i < D; i += blockDim.x) {
        lws[i] = Ws[i];
        lwf[i] = Wf[i];
    }
    __syncthreads();

    const int wave = threadIdx.x >> 5;
    const int lane = threadIdx.x & 31;
    const int tile = blockIdx.x * 8 + wave;
    if (tile >= ntiles) return;  // wave-uniform exit; EXEC stays all-1s

    const int row  = tile * 16 + (lane & 15);
    const int rowc = row < N ? row : N - 1;
    const float* __restrict__ hrow  = h  + (size_t)rowc * D;
    const float* __restrict__ ohrow = oh + (size_t)rowc * D;

    // ISA 7.12.2: A 16x4 f32 -> VGPR0 holds K=0 (lanes 0-15) / K=2 (16-31),
    // VGPR1 holds K=1 / K=3.  Same half-wave split used for B rows.
    const int kA = (lane >> 4) << 1;  // 0 or 2

    v8f acc1 = {};
    v8f acc2 = {};
    for (int kk = 0; kk < D; kk += 4) {
        v2f a1, b1, a2, b2;
        a1.x = hrow[kk + kA];      a1.y = hrow[kk + kA + 1];
        b1.x = lws[kk + kA];       b1.y = lws[kk + kA + 1];
        a2.x = ohrow[kk + kA];     a2.y = ohrow[kk + kA + 1];
        b2.x = lwf[kk + kA];       b2.y = lwf[kk + kA + 1];
        acc1 = __builtin_amdgcn_wmma_f32_16x16x4_f32(
            false, a1, false, b1, (short)0, acc1, false, false);
        acc2 = __builtin_amdgcn_wmma_f32_16x16x4_f32(
            false, a2, false, b2, (short)0, acc2, false, false);
    }

    // softmax over the 2 combination weights (scalar work, every lane)
    float e0 = __expf(wv[0]);
    float e1 = __expf(wv[1]);
    float inv = 1.0f / (e0 + e1);
    float w0 = e0 * inv, w1 = e1 * inv;
    float bsv = bs_p[0], bfv = bf_p[0];

    // C/D layout: lane 0 VGPR m -> (M=m, N=0); lane 16 VGPR m -> (M=8+m, N=0)
    if ((lane & 15) == 0) {
        int rbase = tile * 16 + ((lane >> 4) << 3);
        #pragma unroll
        for (int m = 0; m < 8; ++m) {
            int r = rbase + m;
            if (r < N) {
                float s1 = sigmoidf_dev(acc1[m] + bsv);
                float s2 = sigmoidf_dev(acc2[m] + bfv);
                float sc = w0 * s1 + w1 * s2;
                scores[r] = sc;
                unsigned k = sortkey_desc(sc);
                keys[r] = k;
                atomicAdd(&hist[k >> 16], 1u);
            }
        }
    }
}

// ---------------------------------------------------------------------------
// 2) exclusive scan of 65536-bin histogram (single block, 1024 threads)
// ---------------------------------------------------------------------------
__global__ void scan_kernel(const unsigned* __restrict__ hist,
                            unsigned* __restrict__ offsets) {
    __shared__ unsigned sdata[1024];
    const int t = threadIdx.x;
    const int base = t * (NBINS / 1024);  // 64 bins per thread

    unsigned sum = 0;
    #pragma unroll 4
    for (int b = 0; b < NBINS / 1024; ++b) sum += hist[base + b];
    sdata[t] = sum;
    __syncthreads();

    // Hillis-Steele inclusive scan over per-thread sums
    for (int off = 1; off < 1024; off <<= 1) {
        unsigned v = (t >= off) ? sdata[t - off] : 0u;
        __syncthreads();
        sdata[t] += v;
        __syncthreads();
    }
    unsigned running = sdata[t] - sum;  // exclusive base for this thread
    for (int b = 0; b < NBINS / 1024; ++b) {
        offsets[base + b] = running;
        running += hist[base + b];
    }
    if (t == 1023) offsets[NBINS] = sdata[1023];
}

// ---------------------------------------------------------------------------
// 3) scatter (key, idx) pairs into bin regions
// ---------------------------------------------------------------------------
__global__ void scatter_kernel(const unsigned* __restrict__ keys,
                               const unsigned* __restrict__ offsets,
                               unsigned* __restrict__ binfill,
                               unsigned* __restrict__ skey,
                               int* __restrict__ sidx, int N) {
    int i = blockIdx.x * blockDim.x + threadIdx.x;
    if (i >= N) return;
    unsigned k = keys[i];
    unsigned b = k >> 16;
    unsigned pos = offsets[b] + atomicAdd(&binfill[b], 1u);
    skey[pos] = k;
    sidx[pos] = i;
}

// ---------------------------------------------------------------------------
// 4) per-bin insertion sort (bins average ~1.5 elems) -> exact top_k order:
//    ascending complemented key == descending score; ties -> lower idx first
// ---------------------------------------------------------------------------
__global__ void binsort_kernel(const unsigned* __restrict__ offsets,
                               unsigned* __restrict__ skey,
                               int* __restrict__ sidx) {
    int b = blockIdx.x * blockDim.x + threadIdx.x;
    if (b >= NBINS) return;
    int s = (int)offsets[b];
    int e = (int)offsets[b + 1];
    for (int i = s + 1; i < e; ++i) {
        unsigned k = skey[i];
        int id = sidx[i];
        int j = i - 1;
        while (j >= s && (skey[j] > k || (skey[j] == k && sidx[j] > id))) {
            skey[j + 1] = skey[j];
            sidx[j + 1] = sidx[j];
            --j;
        }
        skey[j + 1] = k;
        sidx[j + 1] = id;
    }
}

// ---------------------------------------------------------------------------
// 5) emit idx output + build remap
// ---------------------------------------------------------------------------
__global__ void emit_kernel(const int* __restrict__ sidx,
                            int* __restrict__ out_idx,
                            int* __restrict__ remap, int SUB) {
    int j = blockIdx.x * blockDim.x + threadIdx.x;
    if (j >= SUB) return;
    int src = sidx[j];
    out_idx[j] = src;
    remap[src] = j;
}

// ---------------------------------------------------------------------------
// 6) gather rows: new_h = h[idx] * score[idx]; new_x = x[idx]   (float4)
// ---------------------------------------------------------------------------
__global__ void gather_kernel(const float* __restrict__ h,
                              const float* __restrict__ x,
                              const float* __restrict__ scores,
                              const int* __restrict__ sidx,
                              float* __restrict__ out_h,
                              float* __restrict__ out_x,
                              int SUB, int D4) {
    const float4* __restrict__ h4 = (const float4*)h;
    const float4* __restrict__ x4 = (const float4*)x;
    float4* __restrict__ oh4 = (float4*)out_h;
    float4* __restrict__ ox4 = (float4*)out_x;
    long long total = (long long)SUB * D4;
    for (long long t = (long long)blockIdx.x * blockDim.x + threadIdx.x;
         t < 2 * total; t += (long long)gridDim.x * blockDim.x) {
        if (t < total) {
            int j = (int)(t / D4);
            int c = (int)(t % D4);
            int src = sidx[j];
            float val = scores[src];
            float4 v = h4[(size_t)src * D4 + c];
            v.x *= val; v.y *= val; v.z *= val; v.w *= val;
            oh4[(size_t)j * D4 + c] = v;
        } else {
            long long t2 = t - total;
            int j = (int)(t2 / D4);
            int c = (int)(t2 % D4);
            int src = sidx[j];
            ox4[(size_t)j * D4 + c] = x4[(size_t)src * D4 + c];
        }
    }
}

// ---------------------------------------------------------------------------
// 7) induced-subgraph edge filter/remap
// ---------------------------------------------------------------------------
__global__ void edge_kernel(const int* __restrict__ ei,
                            const int* __restrict__ remap,
                            int* __restrict__ oe, int E) {
    int e = blockIdx.x * blockDim.x + threadIdx.x;
    if (e >= E) return;
    int rs = remap[ei[e]];
    int rd = remap[ei[E + e]];
    bool keep = (rs >= 0) & (rd >= 0);
    oe[e]     = keep ? rs : -1;
    oe[E + e] = keep ? rd : -1;
}

// ---------------------------------------------------------------------------
extern "C" void kernel_launch(void* const* d_in, const int* in_sizes, int n_in,
                              void* d_out, int out_size, void* d_ws, size_t ws_size,
                              hipStream_t stream) {
    const float* oh = (const float*)d_in[0];
    const float* h  = (const float*)d_in[1];
    const float* x  = (const float*)d_in[2];
    const float* Ws = (const float*)d_in[3];
    const float* bs = (const float*)d_in[4];
    const float* Wf = (const float*)d_in[5];
    const float* bf = (const float*)d_in[6];
    const float* wv = (const float*)d_in[7];
    const int*   ei = (const int*)d_in[8];

    const int D = in_sizes[3];          // W_s is [D,1]
    const int N = in_sizes[1] / D;      // h is [N,D]
    const int E = in_sizes[8] / 2;      // edge_index is [2,E]
    int SUB = N / 2; if (SUB < 2) SUB = 2;
    const int D4 = D / 4;
    const int ntiles = (N + 15) / 16;

    // ---- workspace carve (256B aligned) ----
    char* w = (char*)d_ws;
    size_t off = 0;
    auto carve = [&](size_t bytes) -> char* {
        char* p = w + off;
        off += (bytes + 255) & ~(size_t)255;
        return p;
    };
    unsigned* keys    = (unsigned*)carve((size_t)N * 4);
    float*    scores  = (float*)   carve((size_t)N * 4);
    unsigned* hist    = (unsigned*)carve((size_t)NBINS * 4);
    unsigned* binfill = (unsigned*)carve((size_t)NBINS * 4);
    unsigned* offsets = (unsigned*)carve((size_t)(NBINS + 1) * 4);
    unsigned* skey    = (unsigned*)carve((size_t)N * 4);
    int*      sidx    = (int*)     carve((size_t)N * 4);
    int*      remap   = (int*)     carve((size_t)N * 4);
    (void)ws_size;

    // ---- output carve (tuple flat: new_h, idx, new_x, sub_edge_index) ----
    float* out      = (float*)d_out;
    float* out_newh = out;
    int*   out_idx  = (int*)(out + (size_t)SUB * D);
    float* out_newx = out + (size_t)SUB * D + SUB;
    int*   out_edge = (int*)(out + 2 * (size_t)SUB * D + SUB);
    (void)out_size; (void)n_in;

    // 0) init
    {
        int total = 2 * NBINS + N;
        init_kernel<<<(total + 255) / 256, 256, 0, stream>>>(hist, binfill, remap, N);
    }
    // 1) scores via WMMA
    scores_kernel<<<(ntiles + 7) / 8, 256, 0, stream>>>(
        h, oh, Ws, Wf, bs, bf, wv, scores, keys, hist, N, D, ntiles);
    // 2) scan
    scan_kernel<<<1, 1024, 0, stream>>>(hist, offsets);
    // 3) scatter
    scatter_kernel<<<(N + 255) / 256, 256, 0, stream>>>(keys, offsets, binfill,
                                                        skey, sidx, N);
    // 4) per-bin sort
    binsort_kernel<<<NBINS / 256, 256, 0, stream>>>(offsets, skey, sidx);
    // 5) emit idx + remap
    emit_kernel<<<(SUB + 255) / 256, 256, 0, stream>>>(sidx, out_idx, remap, SUB);
    // 6) gather rows
    {
        long long total = 2LL * SUB * D4;
        int blocks = (int)((total + 255) / 256);
        if (blocks > 65535 * 8) blocks = 65535 * 8;
        gather_kernel<<<blocks, 256, 0, stream>>>(h, x, scores, sidx,
                                                  out_newh, out_newx, SUB, D4);
    }
    // 7) edges
    edge_kernel<<<(E + 255) / 256, 256, 0, stream>>>(ei, remap, out_edge, E);
}